// FNO2D_78975858639406
// MI455X (gfx1250) — compile-verified
//
#include <hip/hip_runtime.h>
#include <hip/hip_bf16.h>
#include <math.h>

// ---------------------------------------------------------------------------
// FNO2D forward for MI455X (gfx1250): wave32 WMMA f32 16x16x4 + TDM async
// tensor loads (tensor_load_to_lds / s_wait_tensorcnt) double-buffering LDS.
//
// Decomposition (exact, because the spectral block is zero outside
// ky in [0..16] U [1008..1023], kx in [0..16]):
//   1) k_coldft : C[f,h,kx] = sum_w u[f,h,w] e^{-2πi kx w/W}      (WMMA + TDM)
//   2) k_rowdft : Uhat[f,ky,kx] = sum_h C[f,h,kx] e^{-2πi ky h/H} (VALU, tiny)
//   3) k_mix    : Y[f,ky,kx] = sum_g A[f,g,ky,kx] Uhat[g,ky,kx]   (VALU, tiny)
//   4) k_invrow : D[f,h,kx] = sum_ky Y e^{+2πi ky h/H} * scale    (VALU, tiny)
//   5) k_fused  : u'[f,h,w] = sum_kx (Dr cos - Di sin)            (WMMA + TDM)
//                           + sum_c conv_w[f,c] u[c,h,w] + b, ReLU
// ---------------------------------------------------------------------------

#define FCH   64
#define HH    1024
#define WWID  1024
#define HW    (HH*WWID)
#define KXN   17      // kx = 0..16
#define KYN   33      // ky = 0..16 and 1008..1023
#define NPAD  48      // 2*KXN = 34 padded to 3 WMMA n-tiles
#define KP    36      // 34 padded to multiple of 4

typedef float v2f __attribute__((ext_vector_type(2)));
typedef float v8f __attribute__((ext_vector_type(8)));
typedef unsigned int u32x4 __attribute__((ext_vector_type(4)));
typedef int i32x4 __attribute__((ext_vector_type(4)));
typedef int i32x8 __attribute__((ext_vector_type(8)));

__device__ __forceinline__ v8f wmma4(v2f a, v2f b, v8f c) {
  // V_WMMA_F32_16X16X4_F32 : D(16x16,f32) = A(16x4) * B(4x16) + C
  return __builtin_amdgcn_wmma_f32_16x16x4_f32(
      false, a, false, b, (short)0, c, false, false);
}

// --- Tensor Data Mover: 2D f32 tile (tile0 x tile1 rows) -> LDS ------------
// D# per CDNA5 ISA.  Groups 2/3 zero (<=2D tensor),
// workgroup_mask = 0 (not in a cluster), no padding / iteration / barrier.
__device__ __forceinline__ void tdm_load_2d(unsigned lds_off,
                                            unsigned long long gaddr,
                                            unsigned tdim0, unsigned tdim1,
                                            unsigned stride0,
                                            unsigned tile0, unsigned tile1) {
  u32x4 g0;
  g0[0] = 1u;                                              // count=1 (valid)
  g0[1] = lds_off;                                         // lds_addr [63:32]
  g0[2] = (unsigned)(gaddr & 0xffffffffull);               // global_addr lo
  g0[3] = (unsigned)((gaddr >> 32) & 0x01ffffffull)        // global_addr hi
          | 0x80000000u;                                   // type=2 [127:126]
  i32x8 g1;
  g1[0] = (int)(2u << 16);                                 // data_size=4B
  g1[1] = (int)((tdim0 & 0xffffu) << 16);                  // tensor_dim0 lo
  g1[2] = (int)(((tdim0 >> 16) & 0xffffu) | ((tdim1 & 0xffffu) << 16));
  g1[3] = (int)(((tdim1 >> 16) & 0xffffu) | ((tile0 & 0xffffu) << 16));
  g1[4] = (int)(tile1 & 0xffffu);                          // tile_dim1, dim2=0
  g1[5] = (int)stride0;                                    // dim0_stride lo
  g1[6] = 0;                                               // stride hi / dim1_stride
  g1[7] = 0;
  i32x4 g2 = {0, 0, 0, 0};
  i32x4 g3 = {0, 0, 0, 0};
#if __clang_major__ >= 23
  i32x8 g4 = {0, 0, 0, 0, 0, 0, 0, 0};
  __builtin_amdgcn_tensor_load_to_lds(g0, g1, g2, g3, g4, 0);
#else
  __builtin_amdgcn_tensor_load_to_lds(g0, g1, g2, g3, 0);
#endif
}

#define C2PI 0.0061359231515425649f   /* 2*pi/1024 */

// --- twiddle tables --------------------------------------------------------
// TW[w][j] : forward,  j=2kx -> cos(2π kx w/W), j=2kx+1 -> -sin(...)  (1024x48)
// TB[j][w] : inverse,  j=2kx -> cos(2π kx w/W), j=2kx+1 -> -sin(...)  (36x1024)
__global__ void k_init_tw(float* __restrict__ TW, float* __restrict__ TB) {
  int id = blockIdx.x * blockDim.x + threadIdx.x;
  if (id < HH * NPAD) {
    int w = id / NPAD, j = id % NPAD;
    float v = 0.f;
    if (j < 2 * KXN) {
      int kx = j >> 1;
      float ph = C2PI * (float)((kx * w) & (WWID - 1));
      v = (j & 1) ? -sinf(ph) : cosf(ph);
    }
    TW[id] = v;
  }
  int id2 = id - HH * NPAD;
  if (id2 >= 0 && id2 < KP * WWID) {
    int j = id2 >> 10, w = id2 & (WWID - 1);
    float v = 0.f;
    if (j < 2 * KXN) {
      int kx = j >> 1;
      float ph = C2PI * (float)((kx * w) & (WWID - 1));
      v = (j & 1) ? -sinf(ph) : cosf(ph);
    }
    TB[id2] = v;
  }
}

// --- encoder: u0[f,p] = enc_b[f] + sum_c enc_w[f,c]*cat(x,u)[c,p] ----------
__global__ void k_encode(const float* __restrict__ uin, const float* __restrict__ xin,
                         const float* __restrict__ ew, const float* __restrict__ eb,
                         float* __restrict__ out) {
  int p = blockIdx.x * blockDim.x + threadIdx.x;
  if (p >= HW) return;
  float i0 = xin[p], i1 = xin[HW + p], i2 = uin[p], i3 = uin[HW + p];
  #pragma unroll 4
  for (int f = 0; f < FCH; ++f) {
    float a = eb[f] + ew[f*4+0]*i0 + ew[f*4+1]*i1 + ew[f*4+2]*i2 + ew[f*4+3]*i3;
    out[(size_t)f * HW + p] = a;
  }
}

// --- stage 1: partial column DFT as WMMA GEMM with TDM double buffering ----
// rows = F*H = 65536, K = 1024, N = 48(padded 34).  block: 128 thr = 4 waves,
// each wave owns 16 rows and all 48 N.
// LDS: A 2x(64x32) via TDM + B 2x(32x48) cooperative = 28 KB.
__global__ void k_coldft(const float* __restrict__ u, const float* __restrict__ TW,
                         float* __restrict__ C2) {
  extern __shared__ float sm[];
  float* sA = sm;           // 2 * 2048
  float* sB = sm + 4096;    // 2 * 1536
  const int t = threadIdx.x;
  const int rowBase = blockIdx.x * 64;
  const int lane = t & 31, wid = t >> 5;
  const int am = lane & 15, hi = lane >> 4, kb = hi << 1;
  const bool wave0 = (__builtin_amdgcn_readfirstlane(t) == 0);

  v8f acc[3];
  #pragma unroll
  for (int nt = 0; nt < 3; ++nt)
    #pragma unroll
    for (int r = 0; r < 8; ++r) acc[nt][r] = 0.f;

  // preload chunk 0
  #pragma unroll
  for (int i = 0; i < 12; ++i) {
    int e = i * 128 + t; int r = e / NPAD, c = e % NPAD;
    sB[e] = TW[r * NPAD + c];
  }
  if (wave0)
    tdm_load_2d((unsigned)(size_t)(void*)sA,
                (unsigned long long)(size_t)(u + (size_t)rowBase * WWID),
                WWID, FCH * HH, WWID, 32, 64);
  __builtin_amdgcn_s_wait_tensorcnt(0);
  __syncthreads();

  for (int ch = 0; ch < 32; ++ch) {
    const int cur = ch & 1, nxt = cur ^ 1;
    if (ch + 1 < 32) {
      const int k1 = (ch + 1) * 32;
      #pragma unroll
      for (int i = 0; i < 12; ++i) {
        int e = i * 128 + t; int r = e / NPAD, c = e % NPAD;
        sB[nxt * 1536 + e] = TW[(k1 + r) * NPAD + c];
      }
      if (wave0)   // TDM streams next A tile while we compute on current one
        tdm_load_2d((unsigned)(size_t)(void*)(sA + nxt * 2048),
                    (unsigned long long)(size_t)(u + (size_t)rowBase * WWID + k1),
                    WWID, FCH * HH, WWID, 32, 64);
    }
    const float* A = sA + cur * 2048;
    const float* B = sB + cur * 1536;
    #pragma unroll
    for (int k = 0; k < 32; k += 4) {
      v2f a; a.x = A[(wid*16 + am)*32 + k + kb];
             a.y = A[(wid*16 + am)*32 + k + kb + 1];
      #pragma unroll
      for (int nt = 0; nt < 3; ++nt) {
        v2f b; b.x = B[(k + kb    )*NPAD + nt*16 + am];
               b.y = B[(k + kb + 1)*NPAD + nt*16 + am];
        acc[nt] = wmma4(a, b, acc[nt]);
      }
    }
    __builtin_amdgcn_s_wait_tensorcnt(0);
    __syncthreads();
  }
  const int rb = rowBase + wid * 16;
  #pragma unroll
  for (int nt = 0; nt < 3; ++nt)
    #pragma unroll
    for (int r = 0; r < 8; ++r)
      C2[(size_t)(rb + r + 8*hi) * NPAD + nt*16 + am] = acc[nt][r];
}

// --- stage 2: row DFT over h for the 33 needed ky --------------------------
__global__ void k_rowdft(const float* __restrict__ C2, float* __restrict__ UH) {
  int id = blockIdx.x * blockDim.x + threadIdx.x;
  if (id >= FCH * KYN * KXN) return;
  int kx = id % KXN; int rem = id / KXN;
  int kyi = rem % KYN; int f = rem / KYN;
  int ky = (kyi < 17) ? kyi : (HH - KYN + kyi);   // 17..32 -> 1008..1023
  const float* Crow = C2 + (size_t)f * HH * NPAD + 2 * kx;
  float ur = 0.f, ui = 0.f;
  for (int h = 0; h < HH; ++h) {
    float cr = Crow[h * NPAD], ci = Crow[h * NPAD + 1];
    float s, c; sincosf(C2PI * (float)((ky * h) & (HH - 1)), &s, &c);
    ur += cr * c + ci * s;          // (cr+i ci)(c - i s)
    ui += ci * c - cr * s;
  }
  UH[id * 2] = ur; UH[id * 2 + 1] = ui;
}

// --- stage 3: per-mode 64x64 channel mixing --------------------------------
__global__ void k_mix(const float* __restrict__ UH,
                      const float* __restrict__ Ap, const float* __restrict__ An,
                      const float* __restrict__ Dp, const float* __restrict__ Dn,
                      float* __restrict__ Y, int lay) {
  int id = blockIdx.x * blockDim.x + threadIdx.x;
  if (id >= FCH * KYN * KXN) return;
  int kx = id % KXN; int rem = id / KXN;
  int kyi = rem % KYN; int f = rem / KYN;
  float yr = 0.f, yi = 0.f;
  for (int g = 0; g < FCH; ++g) {
    float ur = UH[((g * KYN + kyi) * KXN + kx) * 2];
    float ui = UH[((g * KYN + kyi) * KXN + kx) * 2 + 1];
    float ar, ai;
    size_t fg = ((size_t)lay * FCH + f) * FCH + g;
    if (kx == 0) {
      ar = (kyi < 17) ? Dp[fg * 17 + kyi] : Dn[fg * 16 + (kyi - 17)];
      ai = 0.f;
    } else {
      const float* w = (kyi < 17)
        ? Ap + 2 * ((fg * 17 + kyi) * 16 + (kx - 1))
        : An + 2 * ((fg * 16 + (kyi - 17)) * 16 + (kx - 1));
      ar = w[0]; ai = w[1];
    }
    yr += ar * ur - ai * ui;
    yi += ar * ui + ai * ur;
  }
  Y[id * 2] = yr; Y[id * 2 + 1] = yi;
}

// --- stage 4: inverse row transform + fold irfft normalization -------------
__global__ void k_invrow(const float* __restrict__ Y, float* __restrict__ D2) {
  int id = blockIdx.x * blockDim.x + threadIdx.x;
  if (id >= FCH * HH * KXN) return;
  int kx = id % KXN; int rem = id / KXN;
  int h = rem & (HH - 1); int f = rem >> 10;
  float dr = 0.f, di = 0.f;
  for (int kyi = 0; kyi < KYN; ++kyi) {
    int ky = (kyi < 17) ? kyi : (HH - KYN + kyi);
    float yr = Y[((f * KYN + kyi) * KXN + kx) * 2];
    float yi = Y[((f * KYN + kyi) * KXN + kx) * 2 + 1];
    float s, c; sincosf(C2PI * (float)((ky * h) & (HH - 1)), &s, &c);
    dr += yr * c - yi * s;          // (yr+i yi)(c + i s)
    di += yi * c + yr * s;
  }
  float sc = ((kx == 0) ? 1.f : 2.f) * (1.f / ((float)HH * (float)WWID));
  size_t base = ((size_t)f * HH + h) * KP;
  D2[base + 2*kx]     = dr * sc;
  D2[base + 2*kx + 1] = di * sc;
  if (kx == 0) { D2[base + 34] = 0.f; D2[base + 35] = 0.f; }
}

// --- stage 5: fused inverse-w GEMM + 1x1 conv GEMM + bias + ReLU -----------
// block = 256 thr (8 waves); block tile = (one h row) x 128 w x all 64 f.
// wave: 16 w-cols, 4 m-tiles of 16 f.  u-tile (64x128) fetched by TDM.
__global__ void k_fused(const float* __restrict__ ucur, const float* __restrict__ D2,
                        const float* __restrict__ TBg,
                        const float* __restrict__ cw, const float* __restrict__ cb,
                        float* __restrict__ unext, int do_relu) {
  extern __shared__ float sm[];
  float* sConv = sm;             // 64*64   = 4096
  float* sU    = sm + 4096;      // 64*128  = 8192
  float* sD    = sm + 12288;     // 64*36   = 2304
  float* sTB   = sm + 14592;     // 36*128  = 4608   total 19200 floats
  const int t = threadIdx.x;
  const int h = blockIdx.x >> 3;
  const int wbase = (blockIdx.x & 7) << 7;
  const bool wave0 = (__builtin_amdgcn_readfirstlane(t) == 0);

  if (wave0)   // TDM: 64 channel rows x 128 floats, row stride = HW elements
    tdm_load_2d((unsigned)(size_t)(void*)sU,
                (unsigned long long)(size_t)(ucur + (size_t)h * WWID + wbase),
                WWID, FCH, HW, 128, 64);
  #pragma unroll
  for (int i = 0; i < 16; ++i) { int e = i*256 + t; sConv[e] = cw[e]; }
  #pragma unroll
  for (int i = 0; i < 9; ++i) {
    int e = i*256 + t; int f = e / KP, j = e % KP;
    sD[e] = D2[((size_t)f * HH + h) * KP + j];
  }
  #pragma unroll
  for (int i = 0; i < 18; ++i) {
    int e = i*256 + t; int j = e >> 7, w = e & 127;
    sTB[e] = TBg[j * WWID + wbase + w];
  }
  __builtin_amdgcn_s_wait_tensorcnt(0);
  __syncthreads();

  const int lane = t & 31, wv = t >> 5;
  const int am = lane & 15, hi = lane >> 4, kb = hi << 1;
  const int nb = wv * 16 + am;

  v8f acc[4];
  #pragma unroll
  for (int mt = 0; mt < 4; ++mt)
    #pragma unroll
    for (int r = 0; r < 8; ++r) acc[mt][r] = cb[mt*16 + r + 8*hi];

  // conv GEMM: K = 64
  #pragma unroll
  for (int k = 0; k < 64; k += 4) {
    v2f b; b.x = sU[(k + kb) * 128 + nb]; b.y = sU[(k + kb + 1) * 128 + nb];
    #pragma unroll
    for (int mt = 0; mt < 4; ++mt) {
      v2f a; a.x = sConv[(mt*16 + am)*64 + k + kb];
             a.y = sConv[(mt*16 + am)*64 + k + kb + 1];
      acc[mt] = wmma4(a, b, acc[mt]);
    }
  }
  // spectral inverse GEMM: K = 36
  #pragma unroll
  for (int k = 0; k < KP; k += 4) {
    v2f b; b.x = sTB[(k + kb) * 128 + nb]; b.y = sTB[(k + kb + 1) * 128 + nb];
    #pragma unroll
    for (int mt = 0; mt < 4; ++mt) {
      v2f a; a.x = sD[(mt*16 + am)*KP + k + kb];
             a.y = sD[(mt*16 + am)*KP + k + kb + 1];
      acc[mt] = wmma4(a, b, acc[mt]);
    }
  }

  #pragma unroll
  for (int mt = 0; mt < 4; ++mt)
    #pragma unroll
    for (int r = 0; r < 8; ++r) {
      float v = acc[mt][r];
      if (do_relu) v = fmaxf(v, 0.f);
      int f = mt*16 + r + 8*hi;
      unext[(size_t)f * HW + (size_t)h * WWID + wbase + nb] = v;
    }
}

// --- decoder ---------------------------------------------------------------
__global__ void k_decode(const float* __restrict__ u, const float* __restrict__ dw,
                         const float* __restrict__ db, float* __restrict__ out) {
  int p = blockIdx.x * blockDim.x + threadIdx.x;
  if (p >= HW) return;
  float acc = db[0];
  #pragma unroll 8
  for (int f = 0; f < FCH; ++f) acc += dw[f] * u[(size_t)f * HW + p];
  out[p] = acc;
}

// ---------------------------------------------------------------------------
extern "C" void kernel_launch(void* const* d_in, const int* in_sizes, int n_in,
                              void* d_out, int out_size, void* d_ws, size_t ws_size,
                              hipStream_t stream) {
  const float* uin  = (const float*)d_in[0];
  const float* xin  = (const float*)d_in[1];
  const float* ew   = (const float*)d_in[2];
  const float* eb   = (const float*)d_in[3];
  const float* dw   = (const float*)d_in[4];
  const float* db   = (const float*)d_in[5];
  const float* cw   = (const float*)d_in[6];
  const float* cb   = (const float*)d_in[7];
  const float* Ap   = (const float*)d_in[8];   // complex64 interleaved
  const float* An   = (const float*)d_in[9];
  const float* Dp   = (const float*)d_in[10];
  const float* Dn   = (const float*)d_in[11];
  float* out = (float*)d_out;

  float* ws = (float*)d_ws;
  size_t o = 0;
  float* TW = ws + o; o += (size_t)HH * NPAD;              // 49152
  float* TB = ws + o; o += (size_t)KP * WWID;              // 36864
  float* C2 = ws + o; o += (size_t)FCH * HH * NPAD;        // 3145728
  float* UH = ws + o; o += (size_t)FCH * KYN * KXN * 2;    // 71808
  float* Y  = ws + o; o += (size_t)FCH * KYN * KXN * 2;    // 71808
  float* D2 = ws + o; o += (size_t)FCH * HH * KP;          // 2359296
  float* uA = ws + o; o += (size_t)FCH * HW;               // 67108864
  float* uB = ws + o; o += (size_t)FCH * HW;               // 67108864
  if (ws_size < o * sizeof(float)) return;                 // insufficient scratch

  const int nmode = FCH * KYN * KXN;                       // 35904

  k_init_tw<<<(HH*NPAD + KP*WWID + 255) / 256, 256, 0, stream>>>(TW, TB);
  k_encode<<<HW / 256, 256, 0, stream>>>(uin, xin, ew, eb, uA);

  float* ucur = uA; float* unext = uB;
  for (int i = 0; i < 4; ++i) {
    k_coldft<<<(FCH * HH) / 64, 128, (4096 + 3072) * sizeof(float), stream>>>(ucur, TW, C2);
    k_rowdft<<<(nmode + 255) / 256, 256, 0, stream>>>(C2, UH);
    k_mix<<<(nmode + 255) / 256, 256, 0, stream>>>(UH, Ap, An, Dp, Dn, Y, i);
    k_invrow<<<(FCH * HH * KXN + 255) / 256, 256, 0, stream>>>(Y, D2);
    k_fused<<<HH * (WWID / 128), 256, 19200 * sizeof(float), stream>>>(
        ucur, D2, TB, cw + (size_t)i * FCH * FCH, cb + (size_t)i * FCH,
        unext, (i != 3) ? 1 : 0);
    float* tmp = ucur; ucur = unext; unext = tmp;
  }
  k_decode<<<HW / 256, 256, 0, stream>>>(ucur, dw, db, out);
}